// YOLOLoss_46849503265144
// MI455X (gfx1250) — compile-verified
//
#include <hip/hip_runtime.h>

// ---- YOLO constants (from reference) ----
#define SS            7
#define NCELLS        (16384 * 49)       // N * S * S = 802816
#define BLOCK         256
#define NBLOCKS       (NCELLS / BLOCK)   // 3136 exactly -> no tail, EXEC all-ones
#define LAMBDA_COORD  5.0f
#define LAMBDA_NOOBJ  0.5f

typedef float v2f __attribute__((ext_vector_type(2)));
typedef float v8f __attribute__((ext_vector_type(8)));

// Wave32 all-lane sum via V_WMMA_F32_16X16X4_F32.
// A (16x4 f32): lane l (0..15) vgpr0 -> A[l][0], vgpr1 -> A[l][1];
//               lane l (16..31) vgpr0 -> A[l-16][2], vgpr1 -> A[l-16][3].
// Set a = {p, 0}, B = all ones => D[m][n] = p_m + p_{m+16} (indep. of n).
// D vgpr v: lanes 0..15 hold M=v, lanes 16..31 hold M=8+v.
// Sum of 8 D vgprs: lanes 0..15 = sum_{m=0..7}(p_m+p_{m+16}),
//                   lanes 16..31 = sum_{m=8..15}(p_m+p_{m+16}).
// One xor-16 shuffle completes the 32-lane total (broadcast to all lanes).
__device__ __forceinline__ float wave32_sum_wmma(float p) {
    v2f a; a.x = p;    a.y = 0.0f;
    v2f b; b.x = 1.0f; b.y = 1.0f;
    v8f c = {};
    v8f d = __builtin_amdgcn_wmma_f32_16x16x4_f32(
        /*neg_a=*/false, a, /*neg_b=*/false, b,
        /*c_mod=*/(short)0, c, /*reuse_a=*/false, /*reuse_b=*/false);
    float s = ((d[0] + d[1]) + (d[2] + d[3])) + ((d[4] + d[5]) + (d[6] + d[7]));
    s += __shfl_xor(s, 16, 32);
    return s;
}

__global__ __launch_bounds__(BLOCK) void yolo_cell_kernel(
    const float* __restrict__ mo_base,     // (N,7,7,30)
    const float* __restrict__ tg_base,     // (N,7,7,30)
    const float* __restrict__ sizes,       // (N,2) = {H, W}
    float* __restrict__ block_partials)    // NBLOCKS floats
{
    const int cell = blockIdx.x * BLOCK + threadIdx.x;   // 0..NCELLS-1, no tail
    const int n  = cell / 49;
    const int rc = cell - n * 49;
    const int gi = rc / SS;            // row index  -> y_base
    const int gj = rc - gi * SS;       // col index  -> x_base

    // ---- streaming loads: 30 floats per tensor, 8B-aligned (cell*120 B) ----
    float m[30], t[30];
    const v2f* mp = reinterpret_cast<const v2f*>(mo_base + (size_t)cell * 30);
    const v2f* tp = reinterpret_cast<const v2f*>(tg_base + (size_t)cell * 30);
#pragma unroll
    for (int k = 0; k < 15; ++k) {
        v2f a  = __builtin_nontemporal_load(mp + k);   // read-once: NT hint
        v2f bb = __builtin_nontemporal_load(tp + k);
        m[2*k] = a.x;  m[2*k+1] = a.y;
        t[2*k] = bb.x; t[2*k+1] = bb.y;
    }
    const float H = sizes[2*n + 0];
    const float W = sizes[2*n + 1];
    const float gh = H / 7.0f;
    const float gw = W / 7.0f;
    const float xb = (float)gj * gw;
    const float yb = (float)gi * gh;

    // ---- confidence / masks ----
    const float obj    = t[8];                                   // conf_t[...,0]
    const float noobj  = ((t[8] + t[9]) == 0.0f) ? 1.0f : 0.0f;  // sum(conf_t)==0
    const float pc0 = m[8], pc1 = m[9];
    const float sum_pc2 = pc0 * pc0 + pc1 * pc1;

    // ---- target box ----
    const float t_cx = t[0] * gw + xb;
    const float t_cy = t[1] * gh + yb;
    const float t_w  = t[4] * W;
    const float t_h  = t[5] * H;
    const float tx0 = t_cx - 0.5f * t_w, ty0 = t_cy - 0.5f * t_h;
    const float tx1 = t_cx + 0.5f * t_w, ty1 = t_cy + 0.5f * t_h;
    const float area_t = t_w * t_h;

    // ---- IoU for both predicted boxes (branchless) ----
    float iou0, iou1;
#pragma unroll
    for (int b = 0; b < 2; ++b) {
        const float pcx = m[2*b + 0] * gw + xb;
        const float pcy = m[2*b + 1] * gh + yb;
        const float pw  = m[4 + 2*b] * W;
        const float ph  = m[5 + 2*b] * H;
        const float px0 = pcx - 0.5f * pw, py0 = pcy - 0.5f * ph;
        const float px1 = pcx + 0.5f * pw, py1 = pcy + 0.5f * ph;
        const float iw = fmaxf(fminf(tx1, px1) - fmaxf(tx0, px0), 0.0f);
        const float ih = fmaxf(fminf(ty1, py1) - fmaxf(ty0, py0), 0.0f);
        const float inter = iw * ih;
        const float iou_b = inter / (area_t + pw * ph - inter);
        if (b == 0) iou0 = iou_b; else iou1 = iou_b;
    }
    const bool  sel      = (iou1 > iou0);          // jnp.argmax: strict > = first max
    const float iou_best = sel ? iou1 : iou0;
    const float pxy0 = sel ? m[2] : m[0];
    const float pxy1 = sel ? m[3] : m[1];
    const float pwh0 = sel ? m[6] : m[4];
    const float pwh1 = sel ? m[7] : m[5];
    const float pcs  = sel ? pc1  : pc0;

    // ---- loss terms ----
    float loss_class = 0.0f;
#pragma unroll
    for (int k = 0; k < 20; ++k) {
        const float dc = m[10 + k] - t[10 + k];
        loss_class += dc * dc;
    }
    const float dx = t[0] - pxy0, dy = t[1] - pxy1;
    const float loss_xy = LAMBDA_COORD * (dx * dx + dy * dy);
    const float dw = sqrtf(t[4]) - sqrtf(pwh0);
    const float dh = sqrtf(t[5]) - sqrtf(pwh1);
    const float loss_wh = LAMBDA_COORD * (dw * dw + dh * dh);
    const float dco = obj * iou_best - pcs;
    const float loss_conf_obj   = dco * dco;
    const float loss_conf_noobj = LAMBDA_NOOBJ * (sum_pc2 - pcs * pcs);

    const float per_cell = loss_class + loss_xy + loss_wh
                         + loss_conf_obj + loss_conf_noobj;
    const float cell_loss = LAMBDA_NOOBJ * noobj * sum_pc2 + obj * per_cell;

    // ---- reduction: WMMA wave32 sum -> LDS -> per-block partial ----
    const float wsum = wave32_sum_wmma(cell_loss);
    __shared__ float lds[BLOCK / 32];
    const int wave = threadIdx.x >> 5;
    if ((threadIdx.x & 31) == 0) lds[wave] = wsum;
    __syncthreads();
    if (threadIdx.x == 0) {
        float s = 0.0f;
#pragma unroll
        for (int w = 0; w < BLOCK / 32; ++w) s += lds[w];
        block_partials[blockIdx.x] = s;   // deterministic: no atomics
    }
}

__global__ __launch_bounds__(BLOCK) void yolo_final_reduce(
    const float* __restrict__ partials, float* __restrict__ out)
{
    double acc = 0.0;
    for (int idx = threadIdx.x; idx < NBLOCKS; idx += BLOCK)
        acc += (double)partials[idx];
#pragma unroll
    for (int off = 16; off > 0; off >>= 1)
        acc += __shfl_xor(acc, off, 32);
    __shared__ double lds[BLOCK / 32];
    const int wave = threadIdx.x >> 5;
    if ((threadIdx.x & 31) == 0) lds[wave] = acc;
    __syncthreads();
    if (threadIdx.x == 0) {
        double s = 0.0;
#pragma unroll
        for (int w = 0; w < BLOCK / 32; ++w) s += lds[w];
        out[0] = (float)(s * (1.0 / 16384.0));
    }
}

extern "C" void kernel_launch(void* const* d_in, const int* in_sizes, int n_in,
                              void* d_out, int out_size, void* d_ws, size_t ws_size,
                              hipStream_t stream) {
    const float* model_output = (const float*)d_in[0];
    const float* targets      = (const float*)d_in[1];
    const float* sizes        = (const float*)d_in[2];
    float* out      = (float*)d_out;
    float* partials = (float*)d_ws;     // NBLOCKS * 4 B = 12.5 KB scratch

    yolo_cell_kernel<<<NBLOCKS, BLOCK, 0, stream>>>(model_output, targets, sizes, partials);
    yolo_final_reduce<<<1, BLOCK, 0, stream>>>(partials, out);
}